// Encoderdynamic_35579509080228
// MI455X (gfx1250) — compile-verified
//
#include <hip/hip_runtime.h>
#include <stdint.h>

// ---------------- problem constants (from setup_inputs) ----------------
#define B_   32
#define O_   50
#define T_   10647           // 3*(169+676+2704)
#define TB1  507             // 3*169
#define TB2  2535            // 3*(169+676)

typedef __attribute__((ext_vector_type(4))) unsigned int uint32x4_;
typedef __attribute__((ext_vector_type(8))) int          int32x8_;
typedef __attribute__((ext_vector_type(4))) int          int32x4_;

__device__ __forceinline__ float sigmoidf_(float x) {
    return 1.0f / (1.0f + __expf(-x));
}

// =======================================================================
// Kernel A (templated per layer): box decode + IOU-ignore-mask vs GT
// (GT table TDM-staged into LDS) + zero-init of all five target tensors.
// One thread per (b, t) slot of this layer. All divisors compile-time.
// =======================================================================
template<int WGRID, int TBASE>
__global__ void __launch_bounds__(256)
yolo_decode_iou_kernel(const float* __restrict__ outp,   // (B, WGRID, WGRID, 255)
                       const float* __restrict__ anc,    // 6 floats, this layer's anchors
                       const float* __restrict__ gt_boxes,
                       const int*   __restrict__ in_h_p,
                       const int*   __restrict__ in_w_p,
                       float*       __restrict__ out)
{
    constexpr int NT  = WGRID * WGRID * 3;     // slots in this layer per batch
    constexpr int BPB = (NT + 255) / 256;      // blocks per batch

    __shared__ float4 lds_gt[O_];              // 800 B, 16B-aligned for ds_load_b128
    __shared__ float  lds_area[O_];            // precomputed GT areas
    __shared__ float  lds_anc[6];

    const int b     = blockIdx.x / BPB;        // constant divisor
    const int chunk = blockIdx.x - b * BPB;

    if (threadIdx.x < 6) lds_anc[threadIdx.x] = anc[threadIdx.x];

    // ---- Tensor Data Mover: stage gt_boxes[b] (200 f32 = 800 B) into LDS ----
    // Issued by wave 0 only (TDM ignores EXEC but is issued per wave).
    // D#: 2-D tensor, data_size=4 B, tensor_dim0 = tile_dim0 = 200, dim1 = 1.
    if (threadIdx.x < 32) {
        const uint64_t ga  = (uint64_t)(uintptr_t)(gt_boxes + (size_t)b * (O_ * 4));
        const uint32_t lad = (uint32_t)(uintptr_t)(&lds_gt[0]);

        uint32x4_ g0;
        g0.x = 1u;                                               // count=1 (valid user D#)
        g0.y = lad;                                              // lds_addr (bytes)
        g0.z = (uint32_t)(ga & 0xFFFFFFFFu);                     // global_addr[31:0]
        g0.w = (uint32_t)((ga >> 32) & 0x1FFFFFFu) | (2u << 30); // addr[56:32] | type=2

        int32x8_ g1;
        g1.s0 = (int)(2u << 16);     // workgroup_mask=0 | data_size=2 (4 B)
        g1.s1 = (int)(200u << 16);   // atomic_barrier_addr=0 | tensor_dim0[15:0]=200
        g1.s2 = (int)(1u << 16);     // tensor_dim0[31:16]=0  | tensor_dim1[15:0]=1
        g1.s3 = (int)(200u << 16);   // tensor_dim1[31:16]=0  | tile_dim0=200
        g1.s4 = 1;                   // tile_dim1=1 | tile_dim2=0
        g1.s5 = 200;                 // tensor_dim0_stride=200
        g1.s6 = 0;
        g1.s7 = 0;

        int32x4_ gz4 = {0, 0, 0, 0};
        int32x8_ gz8 = {0, 0, 0, 0, 0, 0, 0, 0};

        __builtin_amdgcn_tensor_load_to_lds(g0, g1, gz4, gz4, gz8, 0);
        __builtin_amdgcn_s_wait_tensorcnt(0);
    }
    __syncthreads();

    // precompute GT areas once per block
    if (threadIdx.x < O_) {
        const float4 g = lds_gt[threadIdx.x];
        lds_area[threadIdx.x] = (g.z - g.x) * (g.w - g.y);
    }
    __syncthreads();

    const int t = chunk * 256 + (int)threadIdx.x;   // slot within this layer
    if (t >= NT) return;

    const int f  = t / 3;                  // constant-divisor mul-hi
    const int a  = t - 3 * f;
    const int gy = f / WGRID;              // constant-divisor mul-hi
    const int gx = f - gy * WGRID;

    // gather 4 of 85 channels: outp[b, gy, gx, a*85 + 0..3]
    const size_t cell = (((size_t)b * WGRID + gy) * WGRID + gx) * 255 + (size_t)a * 85;
    const float o0 = outp[cell + 0];
    const float o1 = outp[cell + 1];
    const float o2 = outp[cell + 2];
    const float o3 = outp[cell + 3];

    const int sxi = in_w_p[0] / WGRID;     // python: in_width // w (constant divisor)
    const int syi = in_h_p[0] / WGRID;

    const float px = (sigmoidf_(o0) + (float)gx) * (float)sxi;
    const float py = (sigmoidf_(o1) + (float)gy) * (float)syi;
    const float pw = __expf(o2) * lds_anc[a * 2 + 0];
    const float ph = __expf(o3) * lds_anc[a * 2 + 1];

    const float pl_ = px - 0.5f * pw;
    const float pt_ = py - 0.5f * ph;
    const float pr_ = px + 0.5f * pw;
    const float pb_ = py + 0.5f * ph;
    const float area_p = (pr_ - pl_) * (pb_ - pt_);

    // max(iou) > 0.7  <=>  exists o: inter > 0.7*(area_p + area_g - inter)
    // (denominator > 0 since area_p > 0)  -> no divisions in the hot loop
    bool ignore = false;
#pragma unroll 10
    for (int o = 0; o < O_; ++o) {
        const float4 g  = lds_gt[o];
        const float iw  = fminf(fmaxf(fminf(pr_, g.z) - fmaxf(pl_, g.x), 0.0f), 65504.0f);
        const float ih  = fminf(fmaxf(fminf(pb_, g.w) - fmaxf(pt_, g.y), 0.0f), 65504.0f);
        const float inter = iw * ih;
        ignore = ignore || (inter > 0.7f * (area_p + lds_area[o] - inter));
    }
    const float obj = ignore ? -1.0f : 0.0f;

    // write: zeros everywhere, obj_dyn into obj channel (scatter overrides later)
    const size_t BT = (size_t)B_ * T_;
    const size_t bt = (size_t)b * T_ + (TBASE + t);
    const float2 z2 = make_float2(0.0f, 0.0f);
    *(float2*)(out + bt * 2)          = z2;    // xcyc_t
    *(float2*)(out + 2 * BT + bt * 2) = z2;    // wh_t
    out[4 * BT + bt]                  = obj;   // obj_t
    out[5 * BT + bt]                  = 0.0f;  // cls_t
    *(float2*)(out + 6 * BT + bt * 2) = z2;    // wgt_t
}

// =======================================================================
// Kernel B: per-GT scatter (runs after kernel A on the same stream)
// =======================================================================
__global__ void __launch_bounds__(256)
yolo_scatter_kernel(const int*   __restrict__ matches,
                    const float* __restrict__ anc0,
                    const float* __restrict__ anc1,
                    const float* __restrict__ anc2,
                    const float* __restrict__ gt_boxes,
                    const int*   __restrict__ gt_ids,
                    const int*   __restrict__ in_h_p,
                    const int*   __restrict__ in_w_p,
                    float*       __restrict__ out)
{
    const int idx = blockIdx.x * blockDim.x + threadIdx.x;
    if (idx >= B_ * O_) return;
    const int b = idx / O_;

    const float x1 = gt_boxes[idx * 4 + 0];
    const float y1 = gt_boxes[idx * 4 + 1];
    const float x2 = gt_boxes[idx * 4 + 2];
    const float y2 = gt_boxes[idx * 4 + 3];
    const float w  = x2 - x1, h = y2 - y1;
    const float xc = (x1 + w) * 0.5f;               // faithful: xmax/2
    const float yc = (y1 + h) * 0.5f;
    if ((xc == -1.0f) && (yc == -1.0f) && (w == 0.0f) && (h == 0.0f)) return;

    const int m     = matches[idx];                 // 0..8
    const int nl    = (m < 3) ? 0 : ((m < 6) ? 1 : 2);
    const int grid  = (nl == 0) ? 13 : ((nl == 1) ? 26 : 52);
    const int tbase = (nl == 0) ? 0  : ((nl == 1) ? TB1 : TB2);

    const float in_w = (float)in_w_p[0];
    const float in_h = (float)in_h_p[0];

    const float fx = xc / in_w * (float)grid;
    const float fy = yc / in_h * (float)grid;
    const int   lx = (int)floorf(fx);
    const int   ly = (int)floorf(fy);
    const int   t  = tbase + (ly * grid + lx) * 3 + (m - 3 * nl);

    const float* ap = (m < 3) ? (anc0 + m * 2)
                              : ((m < 6) ? (anc1 + (m - 3) * 2) : (anc2 + (m - 6) * 2));
    const float wv0 = logf(fmaxf(w, 1.0f) / ap[0]);
    const float wv1 = logf(fmaxf(h, 1.0f) / ap[1]);
    const float ws  = 2.0f - w * h / in_w / in_h;

    const size_t BT = (size_t)B_ * T_;
    const size_t bt = (size_t)b * T_ + t;
    out[bt * 2 + 0]          = fx - (float)lx;
    out[bt * 2 + 1]          = fy - (float)ly;
    out[2 * BT + bt * 2 + 0] = wv0;
    out[2 * BT + bt * 2 + 1] = wv1;
    out[4 * BT + bt]         = 1.0f;
    out[5 * BT + bt]         = (float)gt_ids[idx];
    out[6 * BT + bt * 2 + 0] = ws;
    out[6 * BT + bt * 2 + 1] = ws;
}

// =======================================================================
extern "C" void kernel_launch(void* const* d_in, const int* in_sizes, int n_in,
                              void* d_out, int out_size, void* d_ws, size_t ws_size,
                              hipStream_t stream) {
    const int*   matches  = (const int*)  d_in[0];
    // d_in[1] = ious : unused by the reference
    const float* out0     = (const float*)d_in[2];
    const float* out1     = (const float*)d_in[3];
    const float* out2     = (const float*)d_in[4];
    const float* anc0     = (const float*)d_in[5];
    const float* anc1     = (const float*)d_in[6];
    const float* anc2     = (const float*)d_in[7];
    const float* gt_boxes = (const float*)d_in[8];
    const int*   gt_ids   = (const int*)  d_in[9];
    const int*   in_h     = (const int*)  d_in[10];
    const int*   in_w     = (const int*)  d_in[11];
    float*       out      = (float*)d_out;

    // layer 0: 13x13x3 = 507 slots/batch  -> 2 blocks/batch
    yolo_decode_iou_kernel<13, 0><<<dim3(B_ * ((507 + 255) / 256)), dim3(256), 0, stream>>>(
        out0, anc0, gt_boxes, in_h, in_w, out);
    // layer 1: 26x26x3 = 2028 slots/batch -> 8 blocks/batch
    yolo_decode_iou_kernel<26, TB1><<<dim3(B_ * ((2028 + 255) / 256)), dim3(256), 0, stream>>>(
        out1, anc1, gt_boxes, in_h, in_w, out);
    // layer 2: 52x52x3 = 8112 slots/batch -> 32 blocks/batch
    yolo_decode_iou_kernel<52, TB2><<<dim3(B_ * ((8112 + 255) / 256)), dim3(256), 0, stream>>>(
        out2, anc2, gt_boxes, in_h, in_w, out);

    yolo_scatter_kernel<<<dim3((B_ * O_ + 255) / 256), dim3(256), 0, stream>>>(
        matches, anc0, anc1, anc2, gt_boxes, gt_ids, in_h, in_w, out);
}